// VectorQuantizer_66194035966503
// MI455X (gfx1250) — compile-verified
//
#include <hip/hip_runtime.h>
#include <hip/hip_bf16.h>

typedef __attribute__((ext_vector_type(2))) float v2f;
typedef __attribute__((ext_vector_type(8))) float v8f;

#define VQ_D 64
#define VQ_K 1024
#define WAVES_PER_BLOCK 8

// ---------------------------------------------------------------------------
// Kernel A: 0.5 * ||e_k||^2 for each codebook row.
// ---------------------------------------------------------------------------
__global__ void vq_norms_kernel(const float* __restrict__ emb,
                                float* __restrict__ embNormHalf) {
    int k = blockIdx.x * blockDim.x + threadIdx.x;
    if (k < VQ_K) {
        const float* e = emb + k * VQ_D;
        float s = 0.f;
        #pragma unroll
        for (int d = 0; d < VQ_D; ++d) s += e[d] * e[d];
        embNormHalf[k] = 0.5f * s;
    }
}

// ---------------------------------------------------------------------------
// Kernel B: fused score-GEMM (WMMA f32 16x16x4) + argmax + gather + loss part.
// One wave32 handles a 16-row tile of inputs against all 1024 embeddings.
// score = x.e - 0.5*||e||^2  (||x||^2 constant per row -> argmax == argmin dist)
// ---------------------------------------------------------------------------
__global__ void __launch_bounds__(32 * WAVES_PER_BLOCK)
vq_main_kernel(const float* __restrict__ x,
               const float* __restrict__ emb,
               const float* __restrict__ embNormHalf,
               float* __restrict__ out,
               float* __restrict__ partials) {
    const int lane = threadIdx.x & 31;
    const int wave = threadIdx.x >> 5;
    const int tile = blockIdx.x * WAVES_PER_BLOCK + wave; // 16-row tile index
    const int rowBase = tile * 16;
    const int mrow = lane & 15;  // row within tile (A) / embedding within col-tile (B)
    const int half = lane >> 4;  // K-pair selector per WMMA f32 16x16x4 layout

    // Preload A fragments for all 16 k-steps (D=64, K-depth 4 per WMMA).
    // Lane layout: lanes0-15 -> M=lane, VGPR0/1 = K{0,1}; lanes16-31 -> K{2,3}.
    const float* aPtr = x + (size_t)(rowBase + mrow) * VQ_D + half * 2;
    v2f afrag[16];
    #pragma unroll
    for (int k = 0; k < 16; ++k) afrag[k] = *(const v2f*)(aPtr + 4 * k);

    float best[8];
    int   bidx[8];
    #pragma unroll
    for (int i = 0; i < 8; ++i) { best[i] = -3.4e38f; bidx[i] = 0; }

    for (int col = 0; col < VQ_K; col += 16) {
        // C init: -0.5*||e_n||^2, n = col + (lane%16) is constant over all 8 regs.
        const float bias = -embNormHalf[col + mrow];
        v8f acc;
        #pragma unroll
        for (int i = 0; i < 8; ++i) acc[i] = bias;

        // B fragment: B[k][n] = emb[n][k]; same lane addressing pattern as A.
        const float* bPtr = emb + (size_t)(col + mrow) * VQ_D + half * 2;
        #pragma unroll
        for (int k = 0; k < 16; ++k) {
            v2f b = *(const v2f*)(bPtr + 4 * k);
            // args: (neg_a, A, neg_b, B, c_mod, C, reuse_a, reuse_b)
            acc = __builtin_amdgcn_wmma_f32_16x16x4_f32(
                false, afrag[k], false, b, (short)0, acc, false, false);
        }

        // Branch-free running argmax (keeps EXEC all-ones for WMMA legality).
        #pragma unroll
        for (int i = 0; i < 8; ++i) {
            bool p = acc[i] > best[i];
            best[i] = p ? acc[i] : best[i];
            bidx[i] = p ? (col + mrow) : bidx[i];
        }
    }

    // Reduce across the 16 lanes sharing a row (same `half` group).
    // Tie-break to the lowest index == argmin-first semantics of the reference.
    #pragma unroll
    for (int m = 1; m < 16; m <<= 1) {
        #pragma unroll
        for (int i = 0; i < 8; ++i) {
            float ob = __shfl_xor(best[i], m, 32);
            int   oi = __shfl_xor(bidx[i], m, 32);
            bool p = (ob > best[i]) || (ob == best[i] && oi < bidx[i]);
            best[i] = p ? ob : best[i];
            bidx[i] = p ? oi : bidx[i];
        }
    }
    // Now best/bidx[i] uniform within each 16-lane half:
    //   half 0 holds rows M = i, half 1 holds rows M = i + 8.

    // Gather winning codebook rows -> output, fuse (e - x)^2 accumulation.
    float lsum = 0.f;
    #pragma unroll
    for (int r = 0; r < 16; ++r) {
        int idx = __shfl(bidx[r & 7], (r >> 3) << 4, 32); // lane 0 or lane 16
        const float* e  = emb + (size_t)idx * VQ_D;
        const float* xi = x   + (size_t)(rowBase + r) * VQ_D;
        float*       o  = out + (size_t)(rowBase + r) * VQ_D;
        v2f ev = *(const v2f*)(e + lane * 2);
        v2f xv = *(const v2f*)(xi + lane * 2);
        *(v2f*)(o + lane * 2) = ev;
        float d0 = ev.x - xv.x;
        float d1 = ev.y - xv.y;
        lsum += d0 * d0 + d1 * d1;
    }
    #pragma unroll
    for (int m = 1; m < 32; m <<= 1) lsum += __shfl_xor(lsum, m, 32);
    if (lane == 0) partials[tile] = lsum; // deterministic: one writer per slot
}

// ---------------------------------------------------------------------------
// Kernel C: deterministic fixed-order reduction of per-tile partials -> loss.
// loss = (1 + COMMITMENT_COST) * mean((q - x)^2) = 1.25 * sum / (N*D)
// ---------------------------------------------------------------------------
__global__ void vq_loss_kernel(const float* __restrict__ partials, int n,
                               float* __restrict__ lossOut, float denom) {
    __shared__ float s[256];
    float sum = 0.f;
    for (int i = threadIdx.x; i < n; i += 256) sum += partials[i];
    s[threadIdx.x] = sum;
    __syncthreads();
    if (threadIdx.x == 0) {
        float t = 0.f;
        for (int i = 0; i < 256; ++i) t += s[i];
        *lossOut = 1.25f * t / denom;
    }
}

extern "C" void kernel_launch(void* const* d_in, const int* in_sizes, int n_in,
                              void* d_out, int out_size, void* d_ws, size_t ws_size,
                              hipStream_t stream) {
    const float* x   = (const float*)d_in[0];   // [256*512, 64] flat
    const float* emb = (const float*)d_in[1];   // [1024, 64]
    float* out = (float*)d_out;                 // quantized (N*64) ++ loss (1)

    const int nflat = in_sizes[0];              // 8388608
    const int rows  = nflat / VQ_D;             // 131072
    const int tiles = rows / 16;                // 8192

    float* ws = (float*)d_ws;
    float* embNormHalf = ws;                    // 1024 floats
    float* partials    = ws + VQ_K;             // `tiles` floats

    vq_norms_kernel<<<dim3((VQ_K + 255) / 256), dim3(256), 0, stream>>>(
        emb, embNormHalf);

    vq_main_kernel<<<dim3(tiles / WAVES_PER_BLOCK), dim3(32 * WAVES_PER_BLOCK),
                     0, stream>>>(x, emb, embNormHalf, out, partials);

    vq_loss_kernel<<<dim3(1), dim3(256), 0, stream>>>(
        partials, tiles, out + (size_t)rows * VQ_D, (float)nflat);
}